// AllLoss_13400297964003
// MI455X (gfx1250) — compile-verified
//
#include <hip/hip_runtime.h>
#include <math.h>

// ---------------------------------------------------------------------------
// CDNA5 (gfx1250) implementation of the YOLACT-style loss.
// Mask-loss einsum (K=4) mapped onto V_WMMA_F32_16X16X4_F32.
// Hot loop uses hardware TRANS ops (v_exp_f32 / v_log_f32) for softplus.
// ---------------------------------------------------------------------------

typedef __attribute__((ext_vector_type(2))) float v2f;
typedef __attribute__((ext_vector_type(8))) float v8f;

#define NPOS   256          // M
#define NNEG   768          // 3M
#define NGT    32
#define PH     128          // proto H
#define PW     128          // proto W
#define PROTO_HW (PH*PW)    // 16384
#define MASK_HW  (512*512)  // 262144

// accurate version (used only for the 1K-element cls/loc kernel)
__device__ __forceinline__ float softplus_acc(float x) {
    return fmaxf(x, 0.0f) + log1pf(expf(-fabsf(x)));
}

// fast version for the 4.2M-element mask loop: hardware v_exp_f32/v_log_f32
__device__ __forceinline__ float softplus_fast(float x) {
    return fmaxf(x, 0.0f) + __logf(1.0f + __expf(-fabsf(x)));
}

// ---------------------------------------------------------------------------
// Kernel 1: 4x4 sum-pool of gt_masks (32 x 512 x 512) -> pooled (32 x 128 x 128)
// ---------------------------------------------------------------------------
__global__ void pool_masks_kernel(const float* __restrict__ gm,
                                  float* __restrict__ pooled) {
    int idx = blockIdx.x * blockDim.x + threadIdx.x;   // < 32*128*128
    int w = idx & (PW - 1);
    int h = (idx >> 7) & (PH - 1);
    int g = idx >> 14;
    const float* base = gm + ((size_t)g << 18) + (size_t)(h << 2) * 512 + (w << 2);
    float s = 0.0f;
#pragma unroll
    for (int i = 0; i < 4; ++i) {
        float4 row = *reinterpret_cast<const float4*>(base + i * 512);
        s += (row.x + row.y) + (row.z + row.w);
    }
    pooled[idx] = s;
}

// ---------------------------------------------------------------------------
// Kernel 2: classification + localization losses. One block of 256 threads.
// Deterministic LDS tree reduction; result -> clsloc[0].
// ---------------------------------------------------------------------------
__global__ void cls_loc_kernel(const float* __restrict__ map_class,
                               const float* __restrict__ map_box,
                               const float* __restrict__ anchor_center,
                               const float* __restrict__ anchor_box,
                               const float* __restrict__ gt_boxes,
                               const int*   __restrict__ pos_idx,
                               const int*   __restrict__ gt_idx,
                               const int*   __restrict__ neg_idx,
                               float* __restrict__ clsloc) {
    const int m = threadIdx.x;           // 0..255
    float acc = 0.0f;

    // ---- positive sample m: cls + loc ----
    {
        int r  = pos_idx[3 * m + 0];
        int hh = pos_idx[3 * m + 1];
        int ww = pos_idx[3 * m + 2];
        int sp = hh * 64 + ww;

        float logit = map_class[r * 4096 + sp];
        acc += softplus_acc(-logit);

        float a_ch = anchor_center[sp];
        float a_cw = anchor_center[4096 + sp];
        float a_h  = anchor_box[2 * r + 0];
        float a_w  = anchor_box[2 * r + 1];

        int g = gt_idx[m];
        float g0 = gt_boxes[4 * g + 0];
        float g1 = gt_boxes[4 * g + 1];
        float g2 = gt_boxes[4 * g + 2];
        float g3 = gt_boxes[4 * g + 3];

        float tgt0 = (g0 - a_ch) / a_h;
        float tgt1 = (g1 - a_cw) / a_w;
        float tgt2 = log10f(g2 / a_h);
        float tgt3 = log10f(g3 / a_w);
        float tgt[4] = {tgt0, tgt1, tgt2, tgt3};

#pragma unroll
        for (int k = 0; k < 4; ++k) {
            float pred = map_box[(4 * r + k) * 4096 + sp];
            float d  = pred - tgt[k];
            float ad = fabsf(d);
            acc += (ad < 1.0f) ? 0.5f * d * d : (ad - 0.5f);
        }
    }

    // ---- negative samples m, m+256, m+512 ----
#pragma unroll
    for (int t = 0; t < 3; ++t) {
        int j  = m + t * 256;
        int r  = neg_idx[3 * j + 0];
        int hh = neg_idx[3 * j + 1];
        int ww = neg_idx[3 * j + 2];
        float logit = map_class[r * 4096 + hh * 64 + ww];
        acc += softplus_acc(logit);
    }

    __shared__ float red[256];
    red[m] = acc;
    __syncthreads();
#pragma unroll
    for (int s = 128; s > 0; s >>= 1) {
        if (m < s) red[m] += red[m + s];
        __syncthreads();
    }
    if (m == 0) clsloc[0] = red[0];
}

// ---------------------------------------------------------------------------
// Kernel 3: mask loss via V_WMMA_F32_16X16X4_F32.
//   D[m,n] = sum_{p<4} coef[m,p] * proto[p,n]   (m<256, n<16384)
//   contribution per element: 16*softplus(x) - pooled[gt_idx[m]][n] * x
// Each wave: one 16-row m-tile x 4 consecutive 16-col n-tiles (A + gt_idx
// loaded once, 4 WMMAs). 16 m-tiles x 256 n-groups = 4096 waves
// = 512 blocks x 8 waves. Deterministic per-block partials.
// ---------------------------------------------------------------------------
__global__ void
__launch_bounds__(256)
mask_wmma_kernel(const float* __restrict__ proto,      // (4,128,128)
                 const float* __restrict__ map_coef,   // (12,64,64)
                 const int*   __restrict__ pos_idx,    // (256,3)
                 const int*   __restrict__ gt_idx,     // (256,)
                 const float* __restrict__ pooled,     // (32,128,128)
                 float* __restrict__ partials) {
    const int lane   = threadIdx.x & 31;
    const int wave   = threadIdx.x >> 5;
    const int waveId = blockIdx.x * 8 + wave;
    const int mTile  = waveId >> 8;         // 0..15
    const int nGroup = waveId & 255;        // 0..255 -> 4 n-tiles each

    const int mloc  = lane & 15;
    const int khalf = (lane >> 4) << 1;     // 0 for lanes 0-15, 2 for lanes 16-31

    // ---- A fragment: 16x4 coef tile (ISA 32-bit A layout), loaded once ----
    // lane<16: M=lane, K={0,1}; lane>=16: M=lane-16, K={2,3}
    const int m  = mTile * 16 + mloc;
    const int r  = pos_idx[3 * m + 0];
    const int sp = pos_idx[3 * m + 1] * 64 + pos_idx[3 * m + 2];
    v2f a;
    a.x = map_coef[(4 * r + khalf + 0) * 4096 + sp];
    a.y = map_coef[(4 * r + khalf + 1) * 4096 + sp];

    // ---- gt row bases for the 8 D rows this lane touches (hoisted) ----
    // D layout: VGPR j, lane half -> M = j + 8*(lane>=16), N = lane&15
    const int mbase = mTile * 16 + ((lane >> 4) << 3);
    const float* pgrow[8];
#pragma unroll
    for (int j = 0; j < 8; ++j)
        pgrow[j] = pooled + gt_idx[mbase + j] * PROTO_HW;

    const int nBase = nGroup * 64;
    float local = 0.0f;

#pragma unroll
    for (int t = 0; t < 4; ++t) {
        const int n = nBase + t * 16 + mloc;

        // ---- B fragment: 4x16 proto tile ----
        // lane<16: N=lane, K={0,1}; lane>=16: N=lane-16, K={2,3}
        v2f b;
        b.x = proto[(khalf + 0) * PROTO_HW + n];
        b.y = proto[(khalf + 1) * PROTO_HW + n];

        // ---- D = A x B (+0), full f32 WMMA ----
        v8f c = {};
        c = __builtin_amdgcn_wmma_f32_16x16x4_f32(
            /*neg_a=*/false, a, /*neg_b=*/false, b,
            /*c_mod=*/(short)0, c, /*reuse_a=*/false, /*reuse_b=*/false);

        // ---- consume D in-register: BCE against pooled GT ----
#pragma unroll
        for (int j = 0; j < 8; ++j) {
            float x  = c[j];
            float pg = pgrow[j][n];
            local += 16.0f * softplus_fast(x) - pg * x;
        }
    }

    // ---- wave32 butterfly reduce (deterministic) ----
#pragma unroll
    for (int off = 16; off > 0; off >>= 1)
        local += __shfl_xor(local, off, 32);

    __shared__ float wsum[8];
    if (lane == 0) wsum[wave] = local;
    __syncthreads();
    if (threadIdx.x == 0) {
        float s = 0.0f;
#pragma unroll
        for (int i = 0; i < 8; ++i) s += wsum[i];
        partials[blockIdx.x] = s;
    }
}

// ---------------------------------------------------------------------------
// Kernel 4: finalize. Fixed-order sum of 512 mask partials, combine with
// cls+loc, apply mask scale 1/(512^2 * 256).
// ---------------------------------------------------------------------------
__global__ void finalize_kernel(const float* __restrict__ partials,
                                const float* __restrict__ clsloc,
                                float* __restrict__ out) {
    const int t = threadIdx.x;       // 256 threads
    float s = partials[t] + partials[t + 256];
    __shared__ float red[256];
    red[t] = s;
    __syncthreads();
#pragma unroll
    for (int k = 128; k > 0; k >>= 1) {
        if (t < k) red[t] += red[t + k];
        __syncthreads();
    }
    if (t == 0) {
        const float scale = 1.0f / ((float)MASK_HW * (float)NPOS);
        out[0] = clsloc[0] + red[0] * scale;
    }
}

// ---------------------------------------------------------------------------
extern "C" void kernel_launch(void* const* d_in, const int* in_sizes, int n_in,
                              void* d_out, int out_size, void* d_ws, size_t ws_size,
                              hipStream_t stream) {
    (void)in_sizes; (void)n_in; (void)out_size; (void)ws_size;

    const float* proto         = (const float*)d_in[0];
    const float* map_class     = (const float*)d_in[1];
    const float* map_box       = (const float*)d_in[2];
    const float* map_coef      = (const float*)d_in[3];
    const float* anchor_center = (const float*)d_in[4];
    const float* anchor_box    = (const float*)d_in[5];
    const float* gt_boxes      = (const float*)d_in[6];
    const float* gt_masks      = (const float*)d_in[7];
    const int*   pos_idx       = (const int*)d_in[8];
    const int*   gt_idx        = (const int*)d_in[9];
    const int*   neg_idx       = (const int*)d_in[10];
    float* out = (float*)d_out;

    // workspace layout (floats)
    float* ws       = (float*)d_ws;
    float* pooled   = ws;                          // 32*128*128 = 524288 floats
    float* partials = ws + NGT * PROTO_HW;         // 512 floats
    float* clsloc   = partials + 512;              // 1 float

    // 1) pool gt_masks 4x4 -> pooled          (32*128*128 outputs)
    pool_masks_kernel<<<(NGT * PROTO_HW) / 256, 256, 0, stream>>>(gt_masks, pooled);

    // 2) cls + loc losses (single block)
    cls_loc_kernel<<<1, 256, 0, stream>>>(map_class, map_box, anchor_center,
                                          anchor_box, gt_boxes, pos_idx, gt_idx,
                                          neg_idx, clsloc);

    // 3) mask loss GEMM+BCE: 4096 waves x 4 tiles = 512 blocks x 8 waves
    mask_wmma_kernel<<<512, 256, 0, stream>>>(proto, map_coef, pos_idx, gt_idx,
                                              pooled, partials);

    // 4) combine
    finalize_kernel<<<1, 256, 0, stream>>>(partials, clsloc, out);
}